// GCN_25795573580510
// MI455X (gfx1250) — compile-verified
//
#include <hip/hip_runtime.h>

typedef __attribute__((ext_vector_type(2))) float v2f;
typedef __attribute__((ext_vector_type(8))) float v8f;

static __host__ __device__ inline int cdiv(long long a, long long b) { return (int)((a + b - 1) / b); }

// ---------------------------------------------------------------------------
// Degree / normalization kernels
// ---------------------------------------------------------------------------
__global__ void k_fill1(float* __restrict__ p, int n) {
  int i = blockIdx.x * blockDim.x + threadIdx.x;
  if (i < n) p[i] = 1.0f;  // self-loop contributes 1 to in-degree
}

__global__ void k_degcount(const long long* __restrict__ dst, float* __restrict__ deg, int nE) {
  int i = blockIdx.x * blockDim.x + threadIdx.x;
  if (i < nE) atomicAdd(&deg[(int)dst[i]], 1.0f);
}

__global__ void k_rsqrt(float* __restrict__ p, int n) {
  int i = blockIdx.x * blockDim.x + threadIdx.x;
  if (i < n) p[i] = rsqrtf(p[i]);
}

// ---------------------------------------------------------------------------
// WMMA GEMM: C[N x ncols] = A[N x K] * W[K x ncols] (+ bias)
// One block handles 16 rows; wave w handles column tile w (16 cols).
// Uses V_WMMA_F32_16X16X4_F32 (exact f32 matrix pipe).
// ---------------------------------------------------------------------------
template <int K, int NT>
__global__ void gemm16_wmma(const float* __restrict__ A, const float* __restrict__ W,
                            const float* __restrict__ bias, float* __restrict__ C, int ncols) {
  constexpr int NC  = NT * 16;
  constexpr int LDA = K + 4;  // pad: K is a multiple of 64 -> avoid LDS bank conflicts
  __shared__ float sW[K * NC];
  __shared__ float sA[16 * LDA];

  const int tid  = threadIdx.x;
  const int nthr = 32 * NT;
  const int row0 = blockIdx.x * 16;

  // Stage W (zero-padded past ncols) and the 16xK A tile (coalesced) into LDS.
  for (int i = tid; i < K * NC; i += nthr) {
    int kr = i / NC, cc = i - kr * NC;
    sW[i] = (cc < ncols) ? W[kr * ncols + cc] : 0.0f;
  }
  for (int i = tid; i < 16 * K; i += nthr) {
    int r = i / K, cc = i - r * K;
    sA[r * LDA + cc] = A[(long long)row0 * K + i];
  }
  __syncthreads();

  const int wave = tid >> 5;
  const int lane = tid & 31;
  const int half = lane >> 4;  // 0: K pair {0,1}/rows M  |  1: K pair {2,3}
  const int ln   = lane & 15;
  const int kb   = half * 2;

  v8f acc = {};
#pragma unroll
  for (int k = 0; k < K; k += 4) {
    v2f a, b;
    // A 16x4 f32 layout: VGPR0=K(kb), VGPR1=K(kb+1), M = lane&15
    a.x = sA[ln * LDA + k + kb];
    a.y = sA[ln * LDA + k + kb + 1];
    // B 4x16 f32 layout: VGPR0=row kb, VGPR1=row kb+1, N = lane&15
    b.x = sW[(k + kb) * NC + wave * 16 + ln];
    b.y = sW[(k + kb + 1) * NC + wave * 16 + ln];
    acc = __builtin_amdgcn_wmma_f32_16x16x4_f32(false, a, false, b, (short)0, acc, false, false);
  }

  // C/D 16x16 f32 layout: VGPR r -> M = r + 8*half, N = lane&15
#pragma unroll
  for (int r = 0; r < 8; ++r) {
    int m = r + 8 * half;
    int n = wave * 16 + ln;
    if (n < ncols) {
      float v = acc[r];
      if (bias) v += bias[n];
      C[(long long)(row0 + m) * ncols + n] = v;
    }
  }
}

// ---------------------------------------------------------------------------
// GCN aggregation: agg = D^{-1/2}(A+I)D^{-1/2} h  (C = 64 channels)
// ---------------------------------------------------------------------------
__global__ void k_selfinit(const float* __restrict__ h, const float* __restrict__ inv,
                           float* __restrict__ agg, int total /* = N*64 */) {
  int i = blockIdx.x * blockDim.x + threadIdx.x;
  if (i < total) {
    float w = inv[i >> 6];
    agg[i] = h[i] * (w * w);
  }
}

__global__ void k_scatter(const float* __restrict__ h, const long long* __restrict__ src,
                          const long long* __restrict__ dst, const float* __restrict__ inv,
                          float* __restrict__ agg, int nE) {
  long long idx = (long long)blockIdx.x * blockDim.x + threadIdx.x;
  int e = (int)(idx >> 6);
  int c = (int)(idx & 63);
  if (e >= nE) return;
  int s = (int)src[e];
  int d = (int)dst[e];
  float nrm = inv[s] * inv[d];  // edge normalization
  atomicAdd(&agg[(long long)d * 64 + c], h[(long long)s * 64 + c] * nrm);
}

__global__ void k_biasrelu(float* __restrict__ a, const float* __restrict__ b, int total) {
  int i = blockIdx.x * blockDim.x + threadIdx.x;
  if (i < total) {
    float v = a[i] + b[i & 63];
    a[i] = v > 0.0f ? v : 0.0f;
  }
}

// ---------------------------------------------------------------------------
// Launcher
// ---------------------------------------------------------------------------
extern "C" void kernel_launch(void* const* d_in, const int* in_sizes, int n_in,
                              void* d_out, int out_size, void* d_ws, size_t ws_size,
                              hipStream_t stream) {
  const float*     x   = (const float*)d_in[0];
  const long long* ei  = (const long long*)d_in[1];  // int64 [2, E]
  const float*     W1  = (const float*)d_in[2];
  const float*     b1  = (const float*)d_in[3];
  const float*     W2  = (const float*)d_in[4];
  const float*     b2  = (const float*)d_in[5];
  const float*     Wf  = (const float*)d_in[6];
  const float*     bf  = (const float*)d_in[7];

  const int N = in_sizes[0] / 128;  // 100000 (multiple of 16)
  const int E = in_sizes[1] / 2;    // 1600000
  const long long* src = ei;
  const long long* dst = ei + E;

  // Workspace: inv (N f32) | bufH (N*64 f32) | bufA (N*64 f32)
  char* ws    = (char*)d_ws;
  float* inv  = (float*)ws;
  float* bufH = (float*)(ws + (size_t)(1u << 20));
  float* bufA = (float*)(ws + (size_t)(1u << 20) + (size_t)(26u << 20));

  const int NC64   = N * 64;
  const long long edgeWork = (long long)E * 64;

  // 1. inv = rsqrt(in-degree incl. self loop)
  k_fill1<<<cdiv(N, 256), 256, 0, stream>>>(inv, N);
  k_degcount<<<cdiv(E, 256), 256, 0, stream>>>(dst, inv, E);
  k_rsqrt<<<cdiv(N, 256), 256, 0, stream>>>(inv, N);

  // 2. Layer 1: h = x @ W1 ; agg = norm-aggregate ; relu(agg + b1)
  gemm16_wmma<128, 4><<<N / 16, 128, 0, stream>>>(x, W1, nullptr, bufH, 64);
  k_selfinit<<<cdiv(NC64, 256), 256, 0, stream>>>(bufH, inv, bufA, NC64);
  k_scatter<<<cdiv(edgeWork, 256), 256, 0, stream>>>(bufH, src, dst, inv, bufA, E);
  k_biasrelu<<<cdiv(NC64, 256), 256, 0, stream>>>(bufA, b1, NC64);

  // 3. Layer 2: h = relu1 @ W2 ; aggregate ; relu(agg + b2)
  gemm16_wmma<64, 4><<<N / 16, 128, 0, stream>>>(bufA, W2, nullptr, bufH, 64);
  k_selfinit<<<cdiv(NC64, 256), 256, 0, stream>>>(bufH, inv, bufA, NC64);
  k_scatter<<<cdiv(edgeWork, 256), 256, 0, stream>>>(bufH, src, dst, inv, bufA, E);
  k_biasrelu<<<cdiv(NC64, 256), 256, 0, stream>>>(bufA, b2, NC64);

  // 4. Head: out = relu2 @ Wf + bf  (40 cols -> 3 masked col tiles)
  gemm16_wmma<64, 3><<<N / 16, 96, 0, stream>>>(bufA, Wf, bf, (float*)d_out, 40);
}